// WeightedSSIMLoss_24618752541068
// MI455X (gfx1250) — compile-verified
//
#include <hip/hip_runtime.h>
#include <stdint.h>

// ---------------------------------------------------------------------------
// WeightedSSIMLoss for (16,6,384,640) fp32 pairs on gfx1250 (MI455X).
// Bandwidth-bound streaming stencil: async global->LDS row DMA (4-buffer ring,
// 2 rows in flight, 1 barrier/row), separable 3x3 box sums, DSSIM + L1,
// deterministic two-stage reduction. No WMMA by design: 188.7 MB of reads /
// 23.3 TB/s ~= 8.1 us floor, arithmetic (~1 GFLOP) is far below any roof.
// ---------------------------------------------------------------------------

#define Hh        384
#define Ww        640
#define NPLANES   96          // 16 batches * 6 channels
#define STRIPS    8
#define RPS       48          // output rows per strip (last strip: 46)
#define TPB       320         // 2 columns per thread -> covers 640 input cols
#define NBLOCKS   (NPLANES * STRIPS)   // 768

__device__ __forceinline__ float dssim_from_sums(float sx, float sy,
                                                 float sxx, float syy, float sxy) {
  const float inv9 = 1.0f / 9.0f;
  const float K1c  = 1.0e-4f;   // 0.01^2
  const float K2c  = 9.0e-4f;   // 0.03^2
  float mux  = sx * inv9;
  float muy  = sy * inv9;
  float muxx = mux * mux;
  float muyy = muy * muy;
  float muxy = mux * muy;
  float sgx  = fmaf(sxx, inv9, -muxx);
  float sgy  = fmaf(syy, inv9, -muyy);
  float sgxy = fmaf(sxy, inv9, -muxy);
  float num  = fmaf(2.0f, muxy, K1c) * fmaf(2.0f, sgxy, K2c);
  float den  = (muxx + muyy + K1c) * (sgx + sgy + K2c);
  float ssim = num * __builtin_amdgcn_rcpf(den);
  float d    = fmaf(ssim, -0.5f, 0.5f);          // (1 - ssim) * 0.5
  return fminf(fmaxf(d, 0.0f), 1.0f);
}

__global__ void __launch_bounds__(TPB)
ssim_l1_partial(const float* __restrict__ img,
                const float* __restrict__ rec,
                float2* __restrict__ partial) {
  // 4-deep ring of row buffers (async DMA targets) + reduction scratch.
  __shared__ alignas(16) float xrow[4][648];
  __shared__ alignas(16) float yrow[4][648];
  __shared__ float2 red[512];

  const int t     = threadIdx.x;       // owns input cols 2t, 2t+1
  const int strip = blockIdx.x;        // 0..7 row strips
  const int plane = blockIdx.y;        // 0..95

  const int out_lo   = strip * RPS;                            // first output row
  const int out_hi   = (strip == STRIPS - 1) ? 382 : out_lo + RPS; // exclusive
  const int nrows    = out_hi - out_lo + 2;                    // input rows streamed
  const int own_rows = RPS;                                    // L1-owned input rows

  const float* Xp = img + (size_t)plane * (Hh * Ww);
  const float* Yp = rec + (size_t)plane * (Hh * Ww);

  const uint32_t colb  = 8u * (uint32_t)t;              // byte offset of col 2t
  const uint32_t row0b = (uint32_t)out_lo * (uint32_t)(Ww * 4);
  const uint32_t bstride = 648u * 4u;                   // bytes per ring slot

  // Wave-relative LDS byte offsets of slot 0 destinations.
  const uint32_t ldsx0 = (uint32_t)(uintptr_t)&xrow[0][2 * t];
  const uint32_t ldsy0 = (uint32_t)(uintptr_t)&yrow[0][2 * t];
  const float* xbase = &xrow[0][0];
  const float* ybase = &yrow[0][0];

#define ISSUE_ROW(r, buf)                                                       \
  do {                                                                          \
    uint32_t go = row0b + (uint32_t)(r) * (uint32_t)(Ww * 4) + colb;            \
    uint32_t lx = ldsx0 + (uint32_t)(buf) * bstride;                            \
    uint32_t ly = ldsy0 + (uint32_t)(buf) * bstride;                            \
    asm volatile("global_load_async_to_lds_b64 %0, %1, %2 offset:0"             \
                 :: "v"(lx), "v"(go), "s"(Xp) : "memory");                      \
    asm volatile("global_load_async_to_lds_b64 %0, %1, %2 offset:0"             \
                 :: "v"(ly), "v"(go), "s"(Yp) : "memory");                      \
  } while (0)

  float hp1[10], hp2[10];   // ring of horizontal sums, rows r-1 and r-2
#pragma unroll
  for (int i = 0; i < 10; ++i) { hp1[i] = 0.0f; hp2[i] = 0.0f; }

  float ssim_acc = 0.0f;
  float l1_acc   = 0.0f;

  // Two rows of DMA in flight at all times (4 async-loads outstanding).
  ISSUE_ROW(0, 0);
  ISSUE_ROW(1, 1);

  for (int r = 0; r < nrows; ++r) {
    const int nb = r + 2;
    if (nb < nrows) {
      ISSUE_ROW(nb, nb & 3);
      asm volatile("s_wait_asynccnt 4" ::: "memory");  // row r's loads done
    } else if (r + 1 < nrows) {
      asm volatile("s_wait_asynccnt 2" ::: "memory");
    } else {
      asm volatile("s_wait_asynccnt 0" ::: "memory");
    }
    // Single barrier per row: makes row r visible to all waves AND (because it
    // fences the previous iteration's LDS reads) licenses the refill of ring
    // slot (r+2)&3, which was last read at iteration r-2.
    __syncthreads();

    const float* xb = xbase + (r & 3) * 648;
    const float* yb = ybase + (r & 3) * 648;

    // cols 2t..2t+3 (2 halo values come from neighbor thread's DMA)
    const float2 xa = *(const float2*)&xb[2 * t];
    const float2 xc = *(const float2*)&xb[2 * t + 2];
    const float2 ya = *(const float2*)&yb[2 * t];
    const float2 yc = *(const float2*)&yb[2 * t + 2];

    const float a0 = xa.x, a1 = xa.y, a2 = xc.x, a3 = xc.y;
    const float b0 = ya.x, b1 = ya.y, b2 = yc.x, b3 = yc.y;

    // L1 over owned pixels (rows out_lo..out_lo+47, cols 2t,2t+1)
    if (r < own_rows) {
      l1_acc += fabsf(a0 - b0) + fabsf(a1 - b1);
    }

    // per-pixel products
    const float q0 = a0 * a0, q1 = a1 * a1, q2 = a2 * a2, q3 = a3 * a3;
    const float w0 = b0 * b0, w1 = b1 * b1, w2 = b2 * b2, w3 = b3 * b3;
    const float p0 = a0 * b0, p1 = a1 * b1, p2 = a2 * b2, p3 = a3 * b3;

    // horizontal 3-sums: [col][stat], stat = x,y,xx,yy,xy
    float h[10];
    h[0] = a0 + a1 + a2;  h[5] = a1 + a2 + a3;
    h[1] = b0 + b1 + b2;  h[6] = b1 + b2 + b3;
    h[2] = q0 + q1 + q2;  h[7] = q1 + q2 + q3;
    h[3] = w0 + w1 + w2;  h[8] = w1 + w2 + w3;
    h[4] = p0 + p1 + p2;  h[9] = p1 + p2 + p3;

    if (r >= 2 && t < 319) {   // t==319 owns cols 638,639 -> no valid output
      float s[10];
#pragma unroll
      for (int i = 0; i < 10; ++i) s[i] = hp2[i] + hp1[i] + h[i];
      ssim_acc += dssim_from_sums(s[0], s[1], s[2], s[3], s[4]);
      ssim_acc += dssim_from_sums(s[5], s[6], s[7], s[8], s[9]);
    }

#pragma unroll
    for (int i = 0; i < 10; ++i) { hp2[i] = hp1[i]; hp1[i] = h[i]; }
  }
#undef ISSUE_ROW

  // ---- deterministic block reduction (320 -> 1 via padded 512 tree) ----
  red[t] = make_float2(ssim_acc, l1_acc);
  if (t < 192) red[320 + t] = make_float2(0.0f, 0.0f);
  __syncthreads();
#pragma unroll
  for (int sft = 256; sft >= 1; sft >>= 1) {
    if (t < sft) {
      red[t].x += red[t + sft].x;
      red[t].y += red[t + sft].y;
    }
    __syncthreads();
  }
  if (t == 0) partial[blockIdx.y * STRIPS + blockIdx.x] = red[0];
}

__global__ void __launch_bounds__(512)
ssim_l1_final(const float2* __restrict__ partial, float* __restrict__ out) {
  __shared__ float2 arr[512];
  const int t = threadIdx.x;
  float2 v = partial[t];                       // NBLOCKS (768) > 512
  if (t < NBLOCKS - 512) {                     // fold upper 256 partials
    float2 w = partial[t + 512];
    v.x += w.x;
    v.y += w.y;
  }
  arr[t] = v;
  __syncthreads();
#pragma unroll
  for (int s = 256; s >= 1; s >>= 1) {
    if (t < s) {
      arr[t].x += arr[t + s].x;
      arr[t].y += arr[t + s].y;
    }
    __syncthreads();
  }
  if (t == 0) {
    const float N_ssim = 16.0f * 3.0f * 382.0f * 638.0f;  // 11,698,368
    const float N_l1   = 16.0f * 3.0f * 384.0f * 640.0f;  // 11,796,480
    out[0] = 0.85f * (arr[0].x / N_ssim) + 0.15f * (arr[0].y / N_l1);
  }
}

extern "C" void kernel_launch(void* const* d_in, const int* in_sizes, int n_in,
                              void* d_out, int out_size, void* d_ws, size_t ws_size,
                              hipStream_t stream) {
  const float* img = (const float*)d_in[0];   // images (16,6,384,640) fp32
  const float* rec = (const float*)d_in[1];   // recon  (16,6,384,640) fp32
  float2* part = (float2*)d_ws;               // 768 partial (ssim, l1) pairs

  ssim_l1_partial<<<dim3(STRIPS, NPLANES), TPB, 0, stream>>>(img, rec, part);
  ssim_l1_final<<<1, 512, 0, stream>>>(part, (float*)d_out);
}